// Mesh_Renderer_10703058501892
// MI455X (gfx1250) — compile-verified
//
#include <hip/hip_runtime.h>

// ---------------------------------------------------------------------------
// Mesh silhouette renderer for MI455X (gfx1250, wave32).
// Rasterization inner loop expressed as rank-3 (K padded to 4) f32 GEMM via
// V_WMMA_F32_16X16X4_F32:  E[edge_rows x pixels] = Coef[rows x 4] * Pix[4 x pixels]
// ---------------------------------------------------------------------------

typedef __attribute__((ext_vector_type(2))) float v2f;
typedef __attribute__((ext_vector_type(8))) float v8f;

#define B_      8
#define V_      642
#define F_      1280
#define IMG     64
#define NPIX    (IMG * IMG)        // 4096 pixels per batch
#define ROWS    (4 * F_)           // 4 edge-rows per face (3 edges + zero pad) = 5120
#define RT      (ROWS / 16)        // 320 row tiles of 16 edge-rows
#define PT      (NPIX / 16)        // 256 pixel tiles of 16 pixels
#define TAN_T   0.26794919243112270647f   // tan(15 deg)
#define EPSV    1e-8f

// ---- Kernel 1: look_at + perspective projection, one thread per (b, v) ----
__global__ void proj_kernel(const float* __restrict__ verts,
                            const float* __restrict__ viewpts,
                            float* __restrict__ xn,
                            float* __restrict__ yn,
                            float* __restrict__ zn) {
    int idx = blockIdx.x * blockDim.x + threadIdx.x;
    if (idx >= B_ * V_) return;
    int b = idx / V_;

    float ex = viewpts[b * 3 + 0];
    float ey = viewpts[b * 3 + 1];
    float ez = viewpts[b * 3 + 2];

    // z_axis = normalize(-eye)
    float nl = sqrtf(ex * ex + ey * ey + ez * ez) + EPSV;
    float zx = -ex / nl, zy = -ey / nl, zz = -ez / nl;
    // x_axis = normalize(cross(up=(0,1,0), z)) = normalize(z.z, 0, -z.x)
    float nx = sqrtf(zz * zz + zx * zx) + EPSV;
    float xx = zz / nx, xy = 0.0f, xz = -zx / nx;
    // y_axis = normalize(cross(z, x))
    float cy0 = zy * xz - zz * xy;
    float cy1 = zz * xx - zx * xz;
    float cy2 = zx * xy - zy * xx;
    float ny = sqrtf(cy0 * cy0 + cy1 * cy1 + cy2 * cy2) + EPSV;
    float yx = cy0 / ny, yy = cy1 / ny, yz = cy2 / ny;

    float px = verts[idx * 3 + 0] - ex;
    float py = verts[idx * 3 + 1] - ey;
    float pz = verts[idx * 3 + 2] - ez;

    float camx = xx * px + xy * py + xz * pz;
    float camy = yx * px + yy * py + yz * pz;
    float camz = zx * px + zy * py + zz * pz;

    float d = camz * TAN_T + EPSV;
    xn[idx] = camx / d;
    yn[idx] = camy / d;
    zn[idx] = camz;
}

// ---- Kernel 2: edge-function coefficients, one thread per (b, f) ----------
// Each face occupies 4 consecutive rows of the WMMA A-matrix:
//   row = (A, B, C, 0) with e(px,py) = A*px + B*py + C.
// Invisible faces get e0=+1, e1=-1 constants -> never all-same-sign.
__global__ void coef_kernel(const int* __restrict__ faces,
                            const float* __restrict__ xn,
                            const float* __restrict__ yn,
                            const float* __restrict__ zn,
                            float4* __restrict__ coef) {
    int idx = blockIdx.x * blockDim.x + threadIdx.x;
    if (idx >= B_ * F_) return;
    int b = idx / F_;
    int f = idx - b * F_;

    int i0 = faces[f * 3 + 0];
    int i1 = faces[f * 3 + 1];
    int i2 = faces[f * 3 + 2];
    const float* X = xn + b * V_;
    const float* Y = yn + b * V_;
    const float* Z = zn + b * V_;

    float ax = X[i0], ay = Y[i0], az = Z[i0];
    float bx = X[i1], by = Y[i1], bz = Z[i1];
    float cx = X[i2], cy = Y[i2], cz = Z[i2];

    bool vis = (az > 0.0f) && (bz > 0.0f) && (cz > 0.0f);

    float4 r0, r1, r2;
    if (vis) {
        // e0 = (bx-ax)(py-ay) - (by-ay)(px-ax)
        r0 = make_float4(-(by - ay), (bx - ax), (by - ay) * ax - (bx - ax) * ay, 0.0f);
        // e1 = (cx-bx)(py-by) - (cy-by)(px-bx)
        r1 = make_float4(-(cy - by), (cx - bx), (cy - by) * bx - (cx - bx) * by, 0.0f);
        // e2 = (ax-cx)(py-cy) - (ay-cy)(px-cx)
        r2 = make_float4(-(ay - cy), (ax - cx), (ay - cy) * cx - (ax - cx) * cy, 0.0f);
    } else {
        r0 = make_float4(0.0f, 0.0f,  1.0f, 0.0f);
        r1 = make_float4(0.0f, 0.0f, -1.0f, 0.0f);
        r2 = make_float4(0.0f, 0.0f,  0.0f, 0.0f);
    }
    float4* p = coef + ((size_t)b * ROWS + 4 * (size_t)f);
    p[0] = r0;
    p[1] = r1;
    p[2] = r2;
    p[3] = make_float4(0.0f, 0.0f, 0.0f, 0.0f);  // pad row
}

// ---- Kernel 3: WMMA rasterization -----------------------------------------
// One wave32 per 16-pixel tile; loop over 320 row-tiles (= 1280 faces).
// A-matrix layout (16x4 f32): lanes 0-15 hold row (lane&15), K=0..1;
// lanes 16-31 hold the SAME rows, K=2..3  -> per-lane b64 coefficient load.
// B-matrix layout (4x16 f32): lanes 0-15 supply rows K=0(px),K=1(py);
// lanes 16-31 supply rows K=2(1.0),K=3(0.0).
// D tile: lane holds 2 faces (rows 0..7 or 8..15) at pixel column (lane&15).
__global__ void __launch_bounds__(256) render_kernel(const float* __restrict__ coef,
                                                     float* __restrict__ out) {
    const int b    = blockIdx.y;
    const int wave = threadIdx.x >> 5;
    const int lane = threadIdx.x & 31;
    const int tile = blockIdx.x * 8 + wave;     // pixel tile [0, 256)

    const int pcol = lane & 15;
    const int pix  = tile * 16 + pcol;
    const int ixp  = pix & (IMG - 1);
    const int iyp  = pix >> 6;
    const float lx =  2.0f * (ixp + 0.5f) / IMG - 1.0f;
    const float ly = -(2.0f * (iyp + 0.5f) / IMG - 1.0f);

    v2f bm;
    if (lane < 16) { bm.x = lx;   bm.y = ly;   }   // K=0 (px), K=1 (py)
    else           { bm.x = 1.0f; bm.y = 0.0f; }   // K=2 (1),  K=3 (0)

    const float* cb = coef + (size_t)b * ROWS * 4;
    const int half = (lane >> 4) << 1;             // K-half float offset in row

    v8f czero = {0.f, 0.f, 0.f, 0.f, 0.f, 0.f, 0.f, 0.f};
    int acc = 0;

    for (int rt = 0; rt < RT; ++rt) {
        // lanes L and L+16 split row L's 16 bytes -> 256B contiguous per wave
        const float* rp = cb + ((size_t)(rt * 16 + pcol) * 4 + half);
        v2f am = *(const v2f*)rp;

        v8f d = __builtin_amdgcn_wmma_f32_16x16x4_f32(
            /*neg_a=*/false, am, /*neg_b=*/false, bm,
            /*c_mod=*/(short)0, czero, /*reuse_a=*/false, /*reuse_b=*/false);

        // face A: d[0..2], face B: d[4..6] (d[3], d[7] are pad rows)
        float mn0 = fminf(fminf(d[0], d[1]), d[2]);
        float mx0 = fmaxf(fmaxf(d[0], d[1]), d[2]);
        float mn1 = fminf(fminf(d[4], d[5]), d[6]);
        float mx1 = fmaxf(fmaxf(d[4], d[5]), d[6]);
        acc |= (int)((mn0 >= 0.0f) | (mx0 <= 0.0f));
        acc |= (int)((mn1 >= 0.0f) | (mx1 <= 0.0f));
    }

    // combine lane L with lane L+16 (other 2 faces of each tile): SWAPX16
    int other = __builtin_amdgcn_ds_swizzle(acc, 0x401F);
    int any = acc | other;
    if (lane < 16) out[b * NPIX + pix] = any ? 1.0f : 0.0f;
}

// ---------------------------------------------------------------------------
extern "C" void kernel_launch(void* const* d_in, const int* in_sizes, int n_in,
                              void* d_out, int out_size, void* d_ws, size_t ws_size,
                              hipStream_t stream) {
    (void)in_sizes; (void)n_in; (void)out_size; (void)ws_size;

    const float* verts   = (const float*)d_in[0];   // [B,V,3] f32
    const float* viewpts = (const float*)d_in[1];   // [B,3]   f32
    const int*   faces   = (const int*)d_in[2];     // [F,3]   i32
    // d_in[3] = img_size scalar (fixed 64), ignored.

    float* ws = (float*)d_ws;
    float* xn = ws;                         // B*V
    float* yn = xn + B_ * V_;               // B*V
    float* zn = yn + B_ * V_;               // B*V
    float4* coef = (float4*)(zn + B_ * V_); // B*ROWS rows of 16B (16B aligned)

    float* out = (float*)d_out;

    proj_kernel<<<(B_ * V_ + 255) / 256, 256, 0, stream>>>(verts, viewpts, xn, yn, zn);
    coef_kernel<<<(B_ * F_ + 255) / 256, 256, 0, stream>>>(faces, xn, yn, zn, coef);

    dim3 grid(PT / 8, B_);                  // 32 x 8 blocks, 8 waves each
    render_kernel<<<grid, 256, 0, stream>>>((const float*)coef, out);
}